// ChamferLoss_83425444758259
// MI455X (gfx1250) — compile-verified
//
#include <hip/hip_runtime.h>

typedef __attribute__((ext_vector_type(2))) float v2f;
typedef __attribute__((ext_vector_type(8))) float v8f;

// One wave computes, for 16 "query" points, min over all Nr "reference"
// points of squared L2 distance, using V_WMMA_F32_16X16X4_F32 (K=4,
// z-coordinate padded with 0).
//
// Trick: fold the norm terms into the WMMA accumulator.
//   C[m][n] = -(||q_m||^2 + ||r_n||^2) / 2
//   c_out   = <q_m, r_n> + C[m][n] = -d[m][n] / 2
//   min_n d = -2 * max_n c_out      (exact scale; clamp >=0 at the end)
// This removes the post-WMMA pk_add chain entirely: per tile we only build C
// (4 v_pk_fma) and update a running max (v_max3 after unroll fusion).
//
// WMMA f32 16x16x4 layouts (CDNA5 ISA):
//   A (16x4, MxK): lanes 0-15 -> M=lane, regs {K0,K1}; lanes 16-31 -> same M, {K2,K3}
//   B (4x16, KxN): symmetric (N on lanes, K split across halves/regs)
//   C/D (16x16):   lane l holds N=l%16; vgpr r holds M=r+8*(l>=16)
__global__ __launch_bounds__(256) void chamfer_dir_kernel(
    const float* __restrict__ q,     // (B, Nq, 3)
    const float* __restrict__ r,     // (B, Nr, 3)
    float* __restrict__ minout,      // (B, Nq): min_j ||q_i - r_j||^2
    int B, int Nq, int Nr)
{
  const int lane = threadIdx.x & 31;
  const int waveInBlock = threadIdx.x >> 5;
  const int waveId = blockIdx.x * (blockDim.x >> 5) + waveInBlock;
  const int tilesQ = Nq >> 4;
  const int b = waveId / tilesQ;
  const int qt = waveId % tilesQ;
  if (b >= B) return;

  const int idx = lane & 15;
  const bool hi = lane >= 16;

  // ---- Load query tile; pack into WMMA A layout branchlessly ----
  const float* qp = q + ((size_t)b * Nq + (size_t)qt * 16 + idx) * 3;
  const float qx = qp[0], qy = qp[1], qz = qp[2];
  v2f A;
  A.x = hi ? qz : qx;
  A.y = hi ? 0.0f : qy;

  // ||q_idx||^2, computed locally (no shuffle needed).
  const float qq = qx * qx + qy * qy + qz * qz;

  // nqq[i] = -0.5 * ||q_{i + 8*hi}||^2 for the C-tile row this (vgpr,half) owns.
  float nqq[8];
#pragma unroll
  for (int i = 0; i < 8; ++i)
    nqq[i] = -0.5f * __shfl(qq, i + (hi ? 8 : 0), 32);

  float cmax[8];
#pragma unroll
  for (int i = 0; i < 8; ++i) cmax[i] = -3.0e38f;

  const float* rbase = r + (size_t)b * Nr * 3;
  const int tilesR = Nr >> 4;
#pragma unroll 4
  for (int rt = 0; rt < tilesR; ++rt) {
    // Each lane loads all 3 coords of tile point idx = lane%16 (one b96 load).
    const float* rp = rbase + ((size_t)rt * 16 + idx) * 3;
    const float rx = rp[0], ry = rp[1], rz = rp[2];
    const float rr2 = rx * rx + ry * ry + rz * rz;   // ||r_n||^2, n = idx

    v2f Bm;
    Bm.x = hi ? rz : rx;
    Bm.y = hi ? 0.0f : ry;

    // C[m][n] = -(||q_m||^2 + ||r_n||^2)/2
    v8f c;
#pragma unroll
    for (int i = 0; i < 8; ++i) c[i] = fmaf(-0.5f, rr2, nqq[i]);

    // c_out = <q_m, r_n> + C = -d/2
    c = __builtin_amdgcn_wmma_f32_16x16x4_f32(
        /*neg_a=*/false, A, /*neg_b=*/false, Bm,
        /*c_mod=*/(short)0, c, /*reuse_a=*/false, /*reuse_b=*/false);

#pragma unroll
    for (int i = 0; i < 8; ++i) cmax[i] = fmaxf(cmax[i], c[i]);
  }

  // Max across the 16 lanes of each half (all 16 lanes share M per vgpr),
  // then d_min = -2 * c_max, clamped to >= 0 like the reference.
#pragma unroll
  for (int i = 0; i < 8; ++i) {
    float v = cmax[i];
    v = fmaxf(v, __shfl_xor(v, 1, 32));
    v = fmaxf(v, __shfl_xor(v, 2, 32));
    v = fmaxf(v, __shfl_xor(v, 4, 32));
    v = fmaxf(v, __shfl_xor(v, 8, 32));
    cmax[i] = v;
  }

  if (idx == 0) {
    float* outp = minout + (size_t)b * Nq + (size_t)qt * 16 + (hi ? 8 : 0);
#pragma unroll
    for (int i = 0; i < 8; ++i) outp[i] = fmaxf(-2.0f * cmax[i], 0.0f);
  }
}

// Deterministic single-block reduction: fixed per-thread accumulation order
// + LDS tree. out = (sum(p2t)/N + sum(t2p)/M) / B
__global__ __launch_bounds__(256) void chamfer_reduce_kernel(
    const float* __restrict__ p2t, const float* __restrict__ t2p,
    float* __restrict__ out, int B, int N, int M)
{
  __shared__ float s1[256];
  __shared__ float s2[256];
  const int tid = threadIdx.x;
  const int tot1 = B * N, tot2 = B * M;
  float acc1 = 0.0f, acc2 = 0.0f;
  for (int i = tid; i < tot1; i += 256) acc1 += p2t[i];
  for (int i = tid; i < tot2; i += 256) acc2 += t2p[i];
  s1[tid] = acc1;
  s2[tid] = acc2;
  __syncthreads();
  for (int s = 128; s > 0; s >>= 1) {
    if (tid < s) { s1[tid] += s1[tid + s]; s2[tid] += s2[tid + s]; }
    __syncthreads();
  }
  if (tid == 0)
    out[0] = (s1[0] / (float)N + s2[0] / (float)M) / (float)B;
}

extern "C" void kernel_launch(void* const* d_in, const int* in_sizes, int n_in,
                              void* d_out, int out_size, void* d_ws, size_t ws_size,
                              hipStream_t stream) {
  (void)in_sizes; (void)n_in; (void)out_size; (void)ws_size;
  const float* pred   = (const float*)d_in[0];   // (B, N, 3) fp32
  const float* target = (const float*)d_in[1];   // (B, M, 3) fp32
  const int B = 16, N = 4096, M = 4096;

  float* min_p2t = (float*)d_ws;                     // B*N floats
  float* min_t2p = min_p2t + (size_t)B * N;          // B*M floats

  // 1 wave per 16 query points, 8 waves (256 threads) per block.
  const int wavesP = B * (N / 16);
  const int blocksP = (wavesP + 7) / 8;
  const int wavesT = B * (M / 16);
  const int blocksT = (wavesT + 7) / 8;

  chamfer_dir_kernel<<<blocksP, 256, 0, stream>>>(pred, target, min_p2t, B, N, M);
  chamfer_dir_kernel<<<blocksT, 256, 0, stream>>>(target, pred, min_t2p, B, M, N);
  chamfer_reduce_kernel<<<1, 256, 0, stream>>>(min_p2t, min_t2p, (float*)d_out, B, N, M);
}